// GNNLayer_3831110828794
// MI455X (gfx1250) — compile-verified
//
#include <hip/hip_runtime.h>

#define BB 4
#define NN 262144
#define FF 64
#define SS 1024
#define GRID_MAIN 4096
#define NT 4            // 4096 blocks * 4 tiles = 16384 tiles = B*N/64

typedef __attribute__((ext_vector_type(2))) float v2f;
typedef __attribute__((ext_vector_type(8))) float v8f;

// ---------------- workspace layout (float offsets) ----------------
#define OFF_SROW 0
#define OFF_SCOL 262144
#define OFF_CROW 524288
#define OFF_CCOL 528384
#define OFF_M4   532480
#define OFF_BC   548864
#define OFF_GA   548928
#define OFF_RC   549184
#define OFF_CC   811328
#define ZERO_CNT 532480

__global__ void k_zero(float* p, int n) {
    int i = blockIdx.x * blockDim.x + threadIdx.x;
    if (i < n) p[i] = 0.0f;
}

// Fold weights: M_m = W_m @ W_out[64m:64m+64, :], b_comb = b_out + sum_m b_m @ W_out[blk m]
__global__ void k_prep(const float* Ws, const float* bs, const float* Wr, const float* br,
                       const float* Wc, const float* bc, const float* Wg, const float* bg,
                       const float* Wo, const float* bo, float* M4, float* b_comb) {
    const float* Wm[4] = {Ws, Wr, Wc, Wg};
    for (int e = threadIdx.x; e < 4 * 64 * 64; e += blockDim.x) {
        int m = e >> 12, r = (e >> 6) & 63, c = e & 63;
        const float* W = Wm[m];
        float acc = 0.0f;
        for (int k = 0; k < 64; ++k)
            acc += W[r * 64 + k] * Wo[(m * 64 + k) * 64 + c];
        M4[e] = acc;
    }
    if (threadIdx.x < 64) {
        int c = threadIdx.x;
        const float* bv[4] = {bs, br, bc, bg};
        float acc = bo[c];
        for (int m = 0; m < 4; ++m)
            for (int k = 0; k < 64; ++k)
                acc += bv[m][k] * Wo[(m * 64 + k) * 64 + c];
        b_comb[c] = acc;
    }
}

// Scatter pass: one thread per (b*N+n, f4-group). 2MB tables -> L2-resident fp32 atomics.
__global__ void k_scatter(const int* __restrict__ idx, const float* __restrict__ val,
                          float* sums_row, float* sums_col, float* cnt_row, float* cnt_col) {
    int g  = blockIdx.x * blockDim.x + threadIdx.x;   // exactly B*N*16 threads
    int nl = g >> 4;                                  // b*N + n
    int f4 = g & 15;
    int b  = nl >> 18;                                // N = 2^18
    const float4 v = *(const float4*)(val + (size_t)nl * 64 + f4 * 4);
    int i0 = idx[nl * 2 + 0];
    int i1 = idx[nl * 2 + 1];
    float* sr = sums_row + (size_t)((b << 10) + i0) * 64 + f4 * 4;
    float* sc = sums_col + (size_t)((b << 10) + i1) * 64 + f4 * 4;
    unsafeAtomicAdd(sr + 0, v.x); unsafeAtomicAdd(sr + 1, v.y);
    unsafeAtomicAdd(sr + 2, v.z); unsafeAtomicAdd(sr + 3, v.w);
    unsafeAtomicAdd(sc + 0, v.x); unsafeAtomicAdd(sc + 1, v.y);
    unsafeAtomicAdd(sc + 2, v.z); unsafeAtomicAdd(sc + 3, v.w);
    if (f4 == 0) {
        unsafeAtomicAdd(cnt_row + (b << 10) + i0, 1.0f);
        unsafeAtomicAdd(cnt_col + (b << 10) + i1, 1.0f);
    }
}

// Global mean path: g = (sum over segments of sums_row)/N ; glob_add = g@M_glob + b_comb
__global__ void k_glob(const float* sums_row, const float* M4, const float* b_comb,
                       float* glob_add) {
    __shared__ float g0[64];
    int b = blockIdx.x, f = threadIdx.x;
    float acc = 0.0f;
    const float* sr = sums_row + ((size_t)b << 16) + f;
    for (int s = 0; s < SS; ++s) acc += sr[s * 64];
    g0[f] = acc * (1.0f / (float)NN);
    __syncthreads();
    const float* Mg = M4 + 3 * 4096;
    float o = b_comb[f];
    for (int k = 0; k < 64; ++k) o += g0[k] * Mg[k * 64 + f];
    glob_add[b * 64 + f] = o;
}

// Per-segment: row_contrib = mean_row@M_row + glob_add ; col_contrib = mean_col@M_col
__global__ void k_contrib(const float* sums_row, const float* sums_col,
                          const float* cnt_row, const float* cnt_col,
                          const float* M4, const float* glob_add,
                          float* row_contrib, float* col_contrib) {
    __shared__ float mr[64], mc[64];
    int bs = blockIdx.x;       // b*1024 + s
    int b  = bs >> 10;
    int f  = threadIdx.x;
    float icr = 1.0f / (cnt_row[bs] + 1e-9f);
    float icc = 1.0f / (cnt_col[bs] + 1e-9f);
    mr[f] = sums_row[(size_t)bs * 64 + f] * icr;
    mc[f] = sums_col[(size_t)bs * 64 + f] * icc;
    __syncthreads();
    const float* Mr = M4 + 4096;
    const float* Mc = M4 + 2 * 4096;
    float orr = glob_add[b * 64 + f], oc = 0.0f;
    for (int k = 0; k < 64; ++k) {
        orr += mr[k] * Mr[k * 64 + f];
        oc  += mc[k] * Mc[k * 64 + f];
    }
    row_contrib[(size_t)bs * 64 + f] = orr;
    col_contrib[(size_t)bs * 64 + f] = oc;
}

// LDS byte offset of a shared-memory pointer: flat LDS address truncates to the
// wave-relative offset in the low 32 bits (CDNA5 aperture mapping).
__device__ __forceinline__ uint32_t lds_off(const void* p) {
    return (uint32_t)(uintptr_t)p;
}

__device__ __forceinline__ void async_copy_b128(uint32_t dst_lds, const void* src) {
    asm volatile("global_load_async_to_lds_b128 %0, %1, off"
                 :: "v"(dst_lds), "v"(src) : "memory");
}

// Main pass: persistent WGs, double-buffered GLOBAL_LOAD_ASYNC_TO_LDS_B128 tile
// prefetch (ASYNCcnt-tracked), V_WMMA_F32_16X16X4_F32 GEMM, fused gather+lrelu epilogue.
__global__ void __launch_bounds__(256)
k_main(const int* __restrict__ idx, const float* __restrict__ val,
       const float* __restrict__ M4, const float* __restrict__ row_contrib,
       const float* __restrict__ col_contrib, float* __restrict__ out_f) {
    __shared__ __align__(16) float s_val[2][4096];   // 64x64 f32 tile, ping-pong
    __shared__ __align__(16) float s_M[4096];        // M_self
    __shared__ __align__(16) float s_out[4096];      // output staging
    __shared__ int s_idx[128];

    int tid = threadIdx.x;
    int blk = blockIdx.x;

    // M_self once per block (L2-resident, sync path).
    const float4* msrc = (const float4*)M4;
    for (int j = tid; j < 1024; j += 256)
        ((float4*)s_M)[j] = msrc[j];

    // Prologue: async-load tile 0 into buffer 0.
    {
        int tl = blk;
        size_t n0 = ((size_t)(tl >> 12)) * NN + (size_t)(tl & 4095) * 64;
        const float4* vsrc = (const float4*)(val + n0 * 64);
        uint32_t base = lds_off(&s_val[0][0]);
#pragma unroll
        for (int k = 0; k < 4; ++k) {
            int j = tid + k * 256;
            async_copy_b128(base + (uint32_t)j * 16, vsrc + j);
        }
    }

    int lane = tid & 31, wave = tid >> 5;
    int r    = wave * 16 + (lane & 15);   // A-frag row (M)
    int hi2  = (lane >> 4) << 1;          // lanes 16-31 hold K+2,K+3
    int col  = lane & 15;                 // B/C/D column
    int rbase = wave * 16 + ((lane >> 4) << 3);

    for (int it = 0; it < NT; ++it) {
        int tl = it * GRID_MAIN + blk;
        int b  = tl >> 12;
        size_t n0 = (size_t)b * NN + (size_t)(tl & 4095) * 64;   // linear row base
        int p = it & 1;

        if (it + 1 < NT) {
            // Prefetch next tile into the other buffer while this one computes.
            int tl2 = (it + 1) * GRID_MAIN + blk;
            size_t n02 = ((size_t)(tl2 >> 12)) * NN + (size_t)(tl2 & 4095) * 64;
            const float4* vsrc = (const float4*)(val + n02 * 64);
            uint32_t base = lds_off(&s_val[p ^ 1][0]);
#pragma unroll
            for (int k = 0; k < 4; ++k) {
                int j = tid + k * 256;
                async_copy_b128(base + (uint32_t)j * 16, vsrc + j);
            }
            // Async loads retire in order: <=4 outstanding => current tile landed.
            asm volatile("s_wait_asynccnt 0x4" ::: "memory");
        } else {
            asm volatile("s_wait_asynccnt 0x0" ::: "memory");
        }
        __syncthreads();   // buf[p] ready for all waves; s_out free again

        const float* sv = s_val[p];
        v8f acc[4] = {};
        for (int s = 0; s < 16; ++s) {
            int koff = 4 * s + hi2;
            v2f a;
            a.x = sv[r * 64 + koff];
            a.y = sv[r * 64 + koff + 1];
#pragma unroll
            for (int t = 0; t < 4; ++t) {
                v2f bb;
                bb.x = s_M[(koff    ) * 64 + t * 16 + col];
                bb.y = s_M[(koff + 1) * 64 + t * 16 + col];
                acc[t] = __builtin_amdgcn_wmma_f32_16x16x4_f32(
                    false, a, false, bb, (short)0, acc[t], false, false);
            }
        }

        int ridx = 0;
        if (tid < 128) ridx = idx[n0 * 2 + tid];

        // C/D layout: VGPR i -> lanes 0-15: M=i, lanes 16-31: M=i+8
#pragma unroll
        for (int t = 0; t < 4; ++t)
#pragma unroll
            for (int i = 0; i < 8; ++i)
                s_out[(rbase + i) * 64 + t * 16 + col] = acc[t][i];
        if (tid < 128) s_idx[tid] = ridx;
        __syncthreads();

        for (int j = tid; j < 1024; j += 256) {
            int row = j >> 4, f4 = j & 15;
            int i0 = s_idx[2 * row], i1 = s_idx[2 * row + 1];
            const float4 rc = *(const float4*)(row_contrib + (size_t)((b << 10) + i0) * 64 + f4 * 4);
            const float4 cc = *(const float4*)(col_contrib + (size_t)((b << 10) + i1) * 64 + f4 * 4);
            const float4 sv4 = ((const float4*)s_out)[j];
            float4 o;
            o.x = sv4.x + rc.x + cc.x;
            o.y = sv4.y + rc.y + cc.y;
            o.z = sv4.z + rc.z + cc.z;
            o.w = sv4.w + rc.w + cc.w;
            o.x = o.x > 0.0f ? o.x : 0.01f * o.x;
            o.y = o.y > 0.0f ? o.y : 0.01f * o.y;
            o.z = o.z > 0.0f ? o.z : 0.01f * o.z;
            o.w = o.w > 0.0f ? o.w : 0.01f * o.w;
            ((float4*)(out_f + (n0 + (size_t)row) * 64))[f4] = o;
        }
        // next iteration's first barrier protects s_out / s_idx reuse
    }
}

extern "C" void kernel_launch(void* const* d_in, const int* in_sizes, int n_in,
                              void* d_out, int out_size, void* d_ws, size_t ws_size,
                              hipStream_t stream) {
    const int*   idx   = (const int*)d_in[0];
    const float* value = (const float*)d_in[1];
    const float* Ws = (const float*)d_in[2];  const float* bs = (const float*)d_in[3];
    const float* Wr = (const float*)d_in[4];  const float* br = (const float*)d_in[5];
    const float* Wc = (const float*)d_in[6];  const float* bc = (const float*)d_in[7];
    const float* Wg = (const float*)d_in[8];  const float* bg = (const float*)d_in[9];
    const float* Wo = (const float*)d_in[10]; const float* bo = (const float*)d_in[11];

    float* ws = (float*)d_ws;
    float* sums_row = ws + OFF_SROW;
    float* sums_col = ws + OFF_SCOL;
    float* cnt_row  = ws + OFF_CROW;
    float* cnt_col  = ws + OFF_CCOL;
    float* M4       = ws + OFF_M4;
    float* b_comb   = ws + OFF_BC;
    float* glob_add = ws + OFF_GA;
    float* row_c    = ws + OFF_RC;
    float* col_c    = ws + OFF_CC;

    // Tuple output: (index, out) -> bit-copy index first, f32 out after.
    hipMemcpyAsync(d_out, d_in[0], (size_t)BB * NN * 2 * sizeof(int),
                   hipMemcpyDeviceToDevice, stream);
    float* out_f = (float*)d_out + (size_t)BB * NN * 2;

    k_zero<<<(ZERO_CNT + 255) / 256, 256, 0, stream>>>(ws, ZERO_CNT);
    k_prep<<<1, 256, 0, stream>>>(Ws, bs, Wr, br, Wc, bc, Wg, bg, Wo, bo, M4, b_comb);
    k_scatter<<<(BB * NN * 16) / 256, 256, 0, stream>>>(idx, value, sums_row, sums_col,
                                                        cnt_row, cnt_col);
    k_glob<<<BB, 64, 0, stream>>>(sums_row, M4, b_comb, glob_add);
    k_contrib<<<BB * SS, 64, 0, stream>>>(sums_row, sums_col, cnt_row, cnt_col,
                                          M4, glob_add, row_c, col_c);
    k_main<<<GRID_MAIN, 256, 0, stream>>>(idx, value, M4, row_c, col_c, out_f);
}